// SoftSphereMultiModel_71064528880284
// MI455X (gfx1250) — compile-verified
//
#include <hip/hip_runtime.h>
#include <stdint.h>

#define NSPEC 8

typedef __attribute__((ext_vector_type(4))) unsigned int u32x4;
typedef __attribute__((ext_vector_type(4))) int          i32x4;
typedef __attribute__((ext_vector_type(8))) int          i32x8;

// ---- CDNA5 Tensor Data Mover: 1-D dword tile, global -> LDS ---------------
#if defined(__HIP_DEVICE_COMPILE__) && __has_builtin(__builtin_amdgcn_tensor_load_to_lds)
#define HAVE_TDM 1
#else
#define HAVE_TDM 0
#endif

__device__ __forceinline__ void tdm_load_dwords(const void* gsrc, uint32_t lds_byte_off,
                                                uint32_t ndw) {
#if HAVE_TDM
  uint64_t ga = (uint64_t)(uintptr_t)gsrc;
  // D# group0 (ISA 8.3): [1:0]=count=1, [63:32]=lds_addr, [120:64]=global_addr,
  // [127:126]=type=2 ("image")
  u32x4 g0 = { 1u,
               lds_byte_off,
               (uint32_t)ga,
               (uint32_t)((ga >> 32) & 0x01FFFFFFull) | (2u << 30) };
  // D# group1 (ISA 8.4): wg_mask=0, data_size=2 (4B), dims in elements.
  // tensor_dim0 bits[79:48], tile_dim0 bits[127:112], tensor_dim0_stride bits[207:160]
  i32x8 g1 = { (int)(2u << 16),
               (int)((ndw & 0xFFFFu) << 16),   // tensor_dim0[15:0]
               (int)((ndw >> 16) & 0xFFFFu),   // tensor_dim0[31:16], tensor_dim1 lo = 0
               (int)((ndw & 0xFFFFu) << 16),   // tile_dim0
               0,                              // tile_dim1=0, tile_dim2=0 (unused)
               (int)ndw,                       // tensor_dim0_stride lo32
               0, 0 };
  i32x4 zz = { 0, 0, 0, 0 };
#if __clang_major__ >= 23
  i32x8 z8 = { 0, 0, 0, 0, 0, 0, 0, 0 };
  __builtin_amdgcn_tensor_load_to_lds(g0, g1, zz, zz, z8, 0);
#else
  __builtin_amdgcn_tensor_load_to_lds(g0, g1, zz, zz, 0);
#endif
#endif
}

__device__ __forceinline__ void wait_tdm() {
#if HAVE_TDM
#if __has_builtin(__builtin_amdgcn_s_wait_tensorcnt)
  __builtin_amdgcn_s_wait_tensorcnt(0);
#else
  asm volatile("s_wait_tensorcnt 0x0" ::: "memory");
#endif
#endif
}

// ---- fused pair + reduction + finalize kernel ------------------------------
__global__ void __launch_bounds__(256)
soft_sphere_pairs(const float* __restrict__ pos,
                  const float* __restrict__ cell,
                  const float* __restrict__ sigma_m,
                  const float* __restrict__ eps_m,
                  const float* __restrict__ alpha_m,
                  const float* __restrict__ shifts,
                  const int*  __restrict__ mapping,
                  const int*  __restrict__ species,
                  float* __restrict__ out,     // [0]=E, [1..n)=energies, then forces, stress
                  float* __restrict__ ws,      // [0..9]=partials, [12]=block ticket (u32)
                  int n_atoms, int n_pairs) {
  __shared__ float s_raw[3 * 64];   // sigma | eps | alpha (raw matrices, TDM target)
  __shared__ float s_sig[64];       // sigma
  __shared__ float s_isg[64];       // 1/sigma
  __shared__ float s_es[64];        // eps/sigma
  __shared__ float s_ea[64];        // eps/alpha
  __shared__ float s_am1[64];       // alpha-1
  __shared__ float s_cell[9];
  __shared__ float s_red[10];

  const int tid = threadIdx.x;
  float* __restrict__ energies = out + 1;
  float* __restrict__ forces   = out + 1 + n_atoms;

#if HAVE_TDM
  if (tid == 0) {
    uint32_t base = (uint32_t)(uintptr_t)(void*)s_raw;  // flat->LDS: low 32 bits
    tdm_load_dwords(sigma_m, base + 0u,   64u);
    tdm_load_dwords(eps_m,   base + 256u, 64u);
    tdm_load_dwords(alpha_m, base + 512u, 64u);
  }
#else
  if (tid < 64) {
    s_raw[tid]       = sigma_m[tid];
    s_raw[64 + tid]  = eps_m[tid];
    s_raw[128 + tid] = alpha_m[tid];
  }
#endif
  if (tid < 9)  s_cell[tid] = cell[tid];
  if (tid < 10) s_red[tid] = 0.0f;
#if HAVE_TDM
  if (tid == 0) wait_tdm();   // wave 0 drains TENSORcnt before the barrier
#endif
  __syncthreads();

  if (tid < 64) {
    const float sg = s_raw[tid];
    const float ep = s_raw[64 + tid];
    const float al = s_raw[128 + tid];
    s_sig[tid] = sg;
    s_isg[tid] = 1.0f / sg;
    s_es[tid]  = ep / sg;
    s_ea[tid]  = ep / al;
    s_am1[tid] = al - 1.0f;
  }
  __syncthreads();

  const float c00 = s_cell[0], c01 = s_cell[1], c02 = s_cell[2];
  const float c10 = s_cell[3], c11 = s_cell[4], c12 = s_cell[5];
  const float c20 = s_cell[6], c21 = s_cell[7], c22 = s_cell[8];

  float e_acc = 0.0f;
  float s00 = 0, s01 = 0, s02 = 0;
  float s10 = 0, s11 = 0, s12 = 0;
  float s20 = 0, s21 = 0, s22 = 0;

  const int stride = gridDim.x * blockDim.x;
  for (int p = blockIdx.x * blockDim.x + tid; p < n_pairs; p += stride) {
    const int pn = p + stride;
    if (pn < n_pairs) {                     // stream-ahead: global_prefetch_b8
      __builtin_prefetch(&shifts[3 * (size_t)pn], 0, 1);
      __builtin_prefetch(&mapping[pn], 0, 1);
      __builtin_prefetch(&mapping[n_pairs + pn], 0, 1);
    }
    const int ai = mapping[p];
    const int aj = mapping[n_pairs + p];
    const float sx = shifts[3 * (size_t)p + 0];
    const float sy = shifts[3 * (size_t)p + 1];
    const float sz = shifts[3 * (size_t)p + 2];
    // dr = r_j - r_i + S @ cell   (S is a row vector)
    const float dx = pos[3 * aj + 0] - pos[3 * ai + 0] + (sx * c00 + sy * c10 + sz * c20);
    const float dy = pos[3 * aj + 1] - pos[3 * ai + 1] + (sx * c01 + sy * c11 + sz * c21);
    const float dz = pos[3 * aj + 2] - pos[3 * ai + 2] + (sx * c02 + sy * c12 + sz * c22);
    const float r2 = dx * dx + dy * dy + dz * dz;
    const int sp = species[ai] * NSPEC + species[aj];
    const float sig = s_sig[sp];
    if (r2 < sig * sig) {
      const float r    = __builtin_amdgcn_sqrtf(r2);
      const float rinv = __builtin_amdgcn_rcpf(r);
      const float base = 1.0f - r * s_isg[sp];              // in (0,1)
      // base^(alpha-1) via v_log_f32 + v_exp_f32
      const float pm1 = __builtin_amdgcn_exp2f(s_am1[sp] * __builtin_amdgcn_logf(base));
      const float e   = s_ea[sp] * pm1 * base;              // (eps/alpha) * base^alpha
      const float fr  = s_es[sp] * pm1 * rinv;              // f_pair / r
      const float fx = fr * dx, fy = fr * dy, fz = fr * dz;
      const float eh = 0.5f * e;
      atomicAdd(&energies[ai], eh);
      atomicAdd(&energies[aj], eh);
      atomicAdd(&forces[3 * ai + 0],  fx);
      atomicAdd(&forces[3 * ai + 1],  fy);
      atomicAdd(&forces[3 * ai + 2],  fz);
      atomicAdd(&forces[3 * aj + 0], -fx);
      atomicAdd(&forces[3 * aj + 1], -fy);
      atomicAdd(&forces[3 * aj + 2], -fz);
      e_acc += e;
      s00 += dx * fx; s01 += dx * fy; s02 += dx * fz;
      s10 += dy * fx; s11 += dy * fy; s12 += dy * fz;
      s20 += dz * fx; s21 += dz * fy; s22 += dz * fz;
    }
  }

  // wave32 shuffle reduction, then ds_add_f32 into LDS
  float vals[10] = { e_acc, s00, s01, s02, s10, s11, s12, s20, s21, s22 };
  const int lane = tid & 31;
#pragma unroll
  for (int k = 0; k < 10; ++k) {
    float v = vals[k];
#pragma unroll
    for (int off = 16; off > 0; off >>= 1) v += __shfl_xor(v, off, 32);
    if (lane == 0) atomicAdd(&s_red[k], v);
  }
  __syncthreads();

  // one thread publishes block partials, last block finalizes (ticket pattern)
  if (tid == 0) {
#pragma unroll
    for (int k = 0; k < 10; ++k) atomicAdd(&ws[k], s_red[k]);
    __threadfence();
    const unsigned ticket = atomicAdd((unsigned int*)(ws + 12), 1u);
    if (ticket == gridDim.x - 1) {            // all blocks' partials are in L2
      float acc[10];
#pragma unroll
      for (int k = 0; k < 10; ++k) acc[k] = atomicAdd(&ws[k], 0.0f);  // coherent read
      const float det = c00 * (c11 * c22 - c12 * c21)
                      - c01 * (c10 * c22 - c12 * c20)
                      + c02 * (c10 * c21 - c11 * c20);
      const float inv_vol = 1.0f / fabsf(det);
      out[0] = 0.5f * acc[0];                 // energy = 0.5 * sum(e_pair)
      const int sbase = 1 + 4 * n_atoms;      // after energies[n] + forces[3n]
#pragma unroll
      for (int k = 0; k < 9; ++k) out[sbase + k] = -acc[1 + k] * inv_vol;
    }
  }
}

// ---- launch ----------------------------------------------------------------
extern "C" void kernel_launch(void* const* d_in, const int* in_sizes, int n_in,
                              void* d_out, int out_size, void* d_ws, size_t ws_size,
                              hipStream_t stream) {
  (void)n_in; (void)ws_size;
  const float* positions = (const float*)d_in[0];
  const float* cell      = (const float*)d_in[1];
  const float* sigma_m   = (const float*)d_in[2];
  const float* eps_m     = (const float*)d_in[3];
  const float* alpha_m   = (const float*)d_in[4];
  const float* shifts    = (const float*)d_in[5];
  const int*   mapping   = (const int*)d_in[6];
  const int*   species   = (const int*)d_in[7];

  const int n_atoms = in_sizes[0] / 3;
  const int n_pairs = in_sizes[5] / 3;

  float* out = (float*)d_out;
  float* ws  = (float*)d_ws;

  // d_out/d_ws are poisoned by the harness; zero them every call
  // (also resets the block ticket at ws[12]).
  hipMemsetAsync(d_out, 0, (size_t)out_size * sizeof(float), stream);
  hipMemsetAsync(d_ws, 0, 16 * sizeof(float), stream);

  const int block = 256;
  int grid = (n_pairs + block - 1) / block;
  if (grid > 4096) grid = 4096;

  soft_sphere_pairs<<<grid, block, 0, stream>>>(positions, cell, sigma_m, eps_m, alpha_m,
                                                shifts, mapping, species,
                                                out, ws, n_atoms, n_pairs);
}